// DeformConvBlock_36816459661556
// MI455X (gfx1250) — compile-verified
//
#include <hip/hip_runtime.h>
#include <hip/hip_bf16.h>

typedef __attribute__((ext_vector_type(2))) float v2f;
typedef __attribute__((ext_vector_type(8))) float v8f;

#define Bn 4
#define Cn 64
#define OCn 64
#define Hn 128
#define Wn 128
#define HWn (Hn * Wn)
#define KKc 576   // C * 9, conv contraction depth (order kk = t*64 + c)

__device__ __forceinline__ int iclamp(int v, int lo, int hi) {
    return v < lo ? lo : (v > hi ? hi : v);
}

// ---------------------------------------------------------------------------
// Kernel 1: offset conv (18 ch from `offset`) + mask conv (9 ch from `input`)
// as implicit-GEMM with V_WMMA_F32_16X16X4_F32, contraction order kk = t*64+c
// so the tap geometry hoists out of the channel loop (no div/mod in hot loop).
// Weights pair-packed in LDS: element kk, col n at [(kk>>1)*stride + n*2 + (kk&1)]
// -> one 8B-aligned ds_load_b64 yields both K elements a lane needs.
// A (16x4 f32): lane pixel = lane&15, VGPR j holds K = kk0 + 2*(lane>>4) + j.
// D (16x16 f32): col = lane&15, VGPR r -> row r + 8*(lane>>4).
// ---------------------------------------------------------------------------
__global__ __launch_bounds__(128) void conv_off_mask_kernel(
    const float* __restrict__ input, const float* __restrict__ offset,
    const float* __restrict__ off_w, const float* __restrict__ off_b,
    const float* __restrict__ mask_w, const float* __restrict__ mask_b,
    float* __restrict__ off_out, float* __restrict__ mask_out)
{
    __shared__ float sBoff[(KKc / 2) * 36 + 64];   // pair-packed, 18 cols (+pad)
    __shared__ float sBmask[(KKc / 2) * 18 + 64];  // pair-packed, 9 cols (+pad)

    const int tid = threadIdx.x;
    // off_w flat is [n][c][t]; repack to kk = t*64+c pair-packed layout.
    for (int i = tid; i < 18 * KKc; i += 128) {
        int n = i / KKc, src = i % KKc;
        int c = src / 9, t = src % 9;
        int kk = t * 64 + c;
        sBoff[(kk >> 1) * 36 + n * 2 + (kk & 1)] = off_w[i];
    }
    for (int i = tid; i < 9 * KKc; i += 128) {
        int n = i / KKc, src = i % KKc;
        int c = src / 9, t = src % 9;
        int kk = t * 64 + c;
        sBmask[(kk >> 1) * 18 + n * 2 + (kk & 1)] = mask_w[i];
    }
    __syncthreads();

    const int wave = tid >> 5;
    const int lane = tid & 31;
    const int m    = lane & 15;   // pixel within tile / output column
    const int hi   = lane >> 4;

    const int tile = blockIdx.x * 4 + wave;          // B*H*(W/16) = 4096 tiles
    const int b    = tile / (Hn * (Wn / 16));
    const int rem  = tile % (Hn * (Wn / 16));
    const int h    = rem / (Wn / 16);
    const int w0   = (rem % (Wn / 16)) * 16;

    // ---- Pass A: offset source, two N tiles (n = m and n = 16+m) share A ----
    {
        const float* base = offset + (size_t)b * Cn * HWn;
        v8f acc0 = {}, acc1 = {};
        for (int t = 0; t < 9; ++t) {
            const int y  = h + (t / 3) - 1;
            const int x  = w0 + m + (t % 3) - 1;
            const float vsel = (y >= 0 && y < Hn && x >= 0 && x < Wn) ? 1.f : 0.f;
            const float* arow = base + (size_t)(2 * hi) * HWn
                              + iclamp(y, 0, Hn - 1) * Wn + iclamp(x, 0, Wn - 1);
#pragma unroll 4
            for (int c0 = 0; c0 < 64; c0 += 4) {
                const int kk = t * 64 + c0 + 2 * hi;
                v2f a;
                a[0] = arow[(size_t)c0 * HWn] * vsel;
                a[1] = arow[(size_t)(c0 + 1) * HWn] * vsel;
                const int pr = kk >> 1;
                v2f b0 = *(const v2f*)&sBoff[pr * 36 + m * 2];
                v2f b1 = *(const v2f*)&sBoff[pr * 36 + (16 + m) * 2]; // n>=18 -> pad garbage, discarded
                acc0 = __builtin_amdgcn_wmma_f32_16x16x4_f32(
                    false, a, false, b0, (short)0, acc0, false, false);
                acc1 = __builtin_amdgcn_wmma_f32_16x16x4_f32(
                    false, a, false, b1, (short)0, acc1, false, false);
            }
        }
        {
            float bias = off_b[m];
#pragma unroll
            for (int r = 0; r < 8; ++r)
                off_out[((size_t)b * 18 + m) * HWn + h * Wn + w0 + r + 8 * hi] = acc0[r] + bias;
        }
        if (m < 2) {
            int n = 16 + m;
            float bias = off_b[n];
#pragma unroll
            for (int r = 0; r < 8; ++r)
                off_out[((size_t)b * 18 + n) * HWn + h * Wn + w0 + r + 8 * hi] = acc1[r] + bias;
        }
    }

    // ---- Pass B: mask source, one N tile (n = m, valid n < 9) ----
    {
        const float* base = input + (size_t)b * Cn * HWn;
        v8f acc = {};
        for (int t = 0; t < 9; ++t) {
            const int y  = h + (t / 3) - 1;
            const int x  = w0 + m + (t % 3) - 1;
            const float vsel = (y >= 0 && y < Hn && x >= 0 && x < Wn) ? 1.f : 0.f;
            const float* arow = base + (size_t)(2 * hi) * HWn
                              + iclamp(y, 0, Hn - 1) * Wn + iclamp(x, 0, Wn - 1);
#pragma unroll 4
            for (int c0 = 0; c0 < 64; c0 += 4) {
                const int kk = t * 64 + c0 + 2 * hi;
                v2f a;
                a[0] = arow[(size_t)c0 * HWn] * vsel;
                a[1] = arow[(size_t)(c0 + 1) * HWn] * vsel;
                v2f bb = *(const v2f*)&sBmask[(kk >> 1) * 18 + m * 2]; // n>=9 -> garbage, discarded
                acc = __builtin_amdgcn_wmma_f32_16x16x4_f32(
                    false, a, false, bb, (short)0, acc, false, false);
            }
        }
        if (m < 9) {
            float bias = mask_b[m];
#pragma unroll
            for (int r = 0; r < 8; ++r)
                mask_out[((size_t)b * 9 + m) * HWn + h * Wn + w0 + r + 8 * hi] = acc[r] + bias;
        }
    }
}

// ---------------------------------------------------------------------------
// Kernel 2: deformable bilinear sampling + grouped contraction.
// WG = 4 waves = 64 pixels of one row, fixed (batch b, group-pair gp).
// Each lane owns one (pixel, tap): computes bilinear coords + mask-folded
// corner weights ONCE in registers, then gathers 16 channels (stride = HW)
// into the LDS A tile. Contraction: 36 x V_WMMA_F32_16X16X4_F32 against a
// 144x16 block-diagonal weight matrix (2 groups x 72 rows), pair-packed.
// ---------------------------------------------------------------------------
#define ASTR 146   // padded A-tile row stride (even -> 8B-aligned b64 loads)

__global__ __launch_bounds__(128) void deform_gemm_kernel(
    const float* __restrict__ input, const float* __restrict__ weight,
    const float* __restrict__ off_in, const float* __restrict__ mask_in,
    float* __restrict__ out)
{
    __shared__ float sBp[72 * 32 + 32];     // pair-packed 144x16 (+pad)
    __shared__ float sA[4][16 * ASTR];      // per-wave 16 x 144 A tile, padded

    const int tid  = threadIdx.x;
    const int wave = tid >> 5;
    const int lane = tid & 31;

    const int bg    = blockIdx.x >> 8;       // 16 (b, gpair) combos
    const int trem  = blockIdx.x & 255;
    const int b     = bg >> 2;
    const int gpair = bg & 3;                // out channels gpair*16 .. +15
    const int h     = trem >> 1;
    const int w0    = (trem & 1) * 64;
    const int pw    = w0 + wave * 16;        // this wave's 16-pixel base

    // Block-diagonal B: row = half*72 + c*9 + k feeds only n in [half*8, half*8+8)
    for (int i = tid; i < 144 * 16; i += 128) {
        int row = i >> 4, n = i & 15;
        int half = row / 72, r = row % 72;
        int c = r / 9, t = r % 9;
        float v = 0.f;
        if ((n >> 3) == half) {
            int oc = (gpair * 2 + half) * 8 + (n & 7);
            v = weight[(size_t)oc * 72 + c * 9 + t];
        }
        sBp[(row >> 1) * 32 + n * 2 + (row & 1)] = v;
    }

    // Sampling: one (pixel, tap) per lane iteration, 16 channels inner.
    const float* ibase = input + (size_t)(b * Cn + gpair * 16) * HWn;
    for (int pk = lane; pk < 144; pk += 32) {
        const int p = pk / 9, k = pk % 9;
        const int w = pw + p;
        const size_t pix = (size_t)h * Wn + w;
        const float dy = off_in[((size_t)b * 18 + 2 * k) * HWn + pix];
        const float dx = off_in[((size_t)b * 18 + 2 * k + 1) * HWn + pix];
        const float mk = mask_in[((size_t)b * 9 + k) * HWn + pix];
        const float ys = (float)(h + k / 3 - 1) + dy;
        const float xs = (float)(w + k % 3 - 1) + dx;
        const float y0f = floorf(ys), x0f = floorf(xs);
        const float fy = ys - y0f, fx = xs - x0f;
        const int y0 = (int)y0f, x0 = (int)x0f;
        // per-corner validity folded into the weights (branch-free gathers)
        const bool yin0 = (y0 >= 0) && (y0 < Hn);
        const bool yin1 = (y0 >= -1) && (y0 < Hn - 1);
        const bool xin0 = (x0 >= 0) && (x0 < Wn);
        const bool xin1 = (x0 >= -1) && (x0 < Wn - 1);
        const float w00 = (yin0 && xin0) ? (1.f - fy) * (1.f - fx) * mk : 0.f;
        const float w01 = (yin0 && xin1) ? (1.f - fy) * fx * mk : 0.f;
        const float w10 = (yin1 && xin0) ? fy * (1.f - fx) * mk : 0.f;
        const float w11 = (yin1 && xin1) ? fy * fx * mk : 0.f;
        const int y0a = iclamp(y0, 0, Hn - 1), y1a = iclamp(y0 + 1, 0, Hn - 1);
        const int x0a = iclamp(x0, 0, Wn - 1), x1a = iclamp(x0 + 1, 0, Wn - 1);
        const int o00 = y0a * Wn + x0a, o01 = y0a * Wn + x1a;
        const int o10 = y1a * Wn + x0a, o11 = y1a * Wn + x1a;
        float* arow = &sA[wave][p * ASTR];
#pragma unroll 4
        for (int c16 = 0; c16 < 16; ++c16) {
            const float* ch = ibase + (size_t)c16 * HWn;
            float v = w00 * ch[o00] + w01 * ch[o01] + w10 * ch[o10] + w11 * ch[o11];
            arow[(c16 >> 3) * 72 + (c16 & 7) * 9 + k] = v;   // match B row order
        }
    }
    __syncthreads();

    // Contraction: 36 x V_WMMA_F32_16X16X4_F32
    const int m  = lane & 15;
    const int hi = lane >> 4;
    const float* aptr = &sA[wave][m * ASTR];
    v8f acc = {};
    for (int kk0 = 0; kk0 < 144; kk0 += 4) {
        const int kk = kk0 + 2 * hi;
        v2f a  = *(const v2f*)&aptr[kk];                       // ds_load_b64
        v2f bb = *(const v2f*)&sBp[(kk >> 1) * 32 + m * 2];    // ds_load_b64
        acc = __builtin_amdgcn_wmma_f32_16x16x4_f32(
            false, a, false, bb, (short)0, acc, false, false);
    }

    const int oc = gpair * 16 + m;
#pragma unroll
    for (int r = 0; r < 8; ++r)
        out[((size_t)b * OCn + oc) * HWn + (size_t)h * Wn + pw + r + 8 * hi] = acc[r];
}

extern "C" void kernel_launch(void* const* d_in, const int* in_sizes, int n_in,
                              void* d_out, int out_size, void* d_ws, size_t ws_size,
                              hipStream_t stream) {
    const float* input  = (const float*)d_in[0];
    const float* offset = (const float*)d_in[1];
    const float* weight = (const float*)d_in[2];
    const float* off_w  = (const float*)d_in[3];
    const float* off_b  = (const float*)d_in[4];
    const float* mask_w = (const float*)d_in[5];
    const float* mask_b = (const float*)d_in[6];
    float* out = (float*)d_out;

    float* off_buf  = (float*)d_ws;                       // B*18*H*W floats
    float* mask_buf = off_buf + (size_t)Bn * 18 * HWn;    // B*9*H*W floats

    conv_off_mask_kernel<<<1024, 128, 0, stream>>>(
        input, offset, off_w, off_b, mask_w, mask_b, off_buf, mask_buf);
    deform_gemm_kernel<<<4096, 128, 0, stream>>>(
        input, weight, off_buf, mask_buf, out);
}